// LabelSmoothing_15839839387991
// MI455X (gfx1250) — compile-verified
//
#include <hip/hip_runtime.h>
#include <cmath>

typedef __attribute__((ext_vector_type(2))) float v2f;
typedef __attribute__((ext_vector_type(4))) float v4f;
typedef __attribute__((ext_vector_type(8))) float v8f;

#define VOCAB 32000
#define WAVES_PER_BLOCK 4
#define THREADS (WAVES_PER_BLOCK * 32)
#define FLOATS_PER_WAVE (VOCAB / WAVES_PER_BLOCK)   // 8000
#define V4_ITERS 62                                  // 62*128 = 7936 floats via b128
#define TAIL_OFF (V4_ITERS * 128)                    // last 64 floats via b64

// One block per row. Each wave streams its 8000-float slice through
// v_wmma_f32_16x16x4_f32 with A = ones (64 f32 folded per wmma).
// b128 loads: one 16B load per lane feeds two wmmas (128 floats/wave/iter).
// Accumulator holds 16 identical copies of the slice sum -> divide by 16.
__global__ __launch_bounds__(THREADS)
void LabelSmoothing_row_kernel(const float* __restrict__ x,
                               const int* __restrict__ target,
                               float* __restrict__ row_terms,
                               float conf, float sv, float c1)
{
    const int row  = blockIdx.x;
    const int lane = threadIdx.x & 31;
    const int wave = threadIdx.x >> 5;

    const float* wbase = x + (size_t)row * VOCAB + wave * FLOATS_PER_WAVE;

    v2f a; a[0] = 1.0f; a[1] = 1.0f;   // A = 16x4 ones
    v8f c = {};

    const float* p = wbase + lane * 4;
#pragma unroll 2
    for (int i = 0; i < V4_ITERS; ++i) {
        v4f v = *(const v4f*)(p + i * 128);        // global_load_b128, coalesced
        v2f b0; b0[0] = v[0]; b0[1] = v[1];
        v2f b1; b1[0] = v[2]; b1[1] = v[3];
        c = __builtin_amdgcn_wmma_f32_16x16x4_f32(
                false, a, false, b0, (short)0, c, false, false);
        c = __builtin_amdgcn_wmma_f32_16x16x4_f32(
                false, a, false, b1, (short)0, c, false, false);
    }
    {   // tail: remaining 64 floats of this wave's slice, one b64 + one wmma
        v2f b = *(const v2f*)(wbase + TAIL_OFF + lane * 2);
        c = __builtin_amdgcn_wmma_f32_16x16x4_f32(
                false, a, false, b, (short)0, c, false, false);
    }

    // Sum all 8 accumulator VGPRs in this lane, then reduce across the wave32.
    float l = c[0] + c[1] + c[2] + c[3] + c[4] + c[5] + c[6] + c[7];
#pragma unroll
    for (int off = 16; off > 0; off >>= 1)
        l += __shfl_xor(l, off, 32);

    __shared__ float wsum[WAVES_PER_BLOCK];
    if (lane == 0) wsum[wave] = l;
    __syncthreads();

    if (threadIdx.x == 0) {
        float s16 = 0.0f;
        for (int w = 0; w < WAVES_PER_BLOCK; ++w) s16 += wsum[w];
        const float S = s16 * (1.0f / 16.0f);   // 16 redundant copies per wmma acc

        const int tgt = target[row];
        float term = 0.0f;
        if (tgt != 0) {                          // padded positions contribute 0
            const float x0 = x[(size_t)row * VOCAB];
            const float xt = x[(size_t)row * VOCAB + tgt];
            term = c1 - conf * xt - sv * (S - x0 - xt);
        }
        row_terms[row] = term;
    }
}

// Deterministic single-block fold of per-row terms into the scalar loss.
__global__ __launch_bounds__(256)
void LabelSmoothing_final_kernel(const float* __restrict__ row_terms,
                                 float* __restrict__ out, int rows)
{
    const int t = threadIdx.x;
    float s = 0.0f;
    for (int i = t; i < rows; i += 256) s += row_terms[i];
#pragma unroll
    for (int off = 16; off > 0; off >>= 1)
        s += __shfl_xor(s, off, 32);

    __shared__ float wsum[8];
    if ((t & 31) == 0) wsum[t >> 5] = s;
    __syncthreads();
    if (t == 0) {
        float tot = 0.0f;
        for (int w = 0; w < 8; ++w) tot += wsum[w];
        out[0] = tot;
    }
}

extern "C" void kernel_launch(void* const* d_in, const int* in_sizes, int n_in,
                              void* d_out, int out_size, void* d_ws, size_t ws_size,
                              hipStream_t stream) {
    const float* x      = (const float*)d_in[0];   // [B,S,V] f32 log-probs
    const int*   target = (const int*)d_in[1];     // [B,S] int
    float* out       = (float*)d_out;              // scalar loss
    float* row_terms = (float*)d_ws;               // rows floats of scratch

    const int rows = in_sizes[1];                  // B*S = 2048

    // Host-side constants in double precision.
    const double conf = 0.9;
    const double sv   = 0.1 / (double)(VOCAB - 2);
    const double c1   = conf * log(conf) + sv * (double)(VOCAB - 2) * log(sv);

    LabelSmoothing_row_kernel<<<rows, THREADS, 0, stream>>>(
        x, target, row_terms, (float)conf, (float)sv, (float)c1);
    LabelSmoothing_final_kernel<<<1, 256, 0, stream>>>(row_terms, out, rows);
}